// MultiHashCodebookLayer_54039278518743
// MI455X (gfx1250) — compile-verified
//
#include <hip/hip_runtime.h>
#include <hip/hip_bf16.h>

typedef __attribute__((ext_vector_type(2))) float v2f;
typedef __attribute__((ext_vector_type(4))) float v4f;
typedef __attribute__((ext_vector_type(8))) float v8f;

#define B_SZ   4096
#define F_SZ   20
#define D_SZ   16
#define H_SZ   2
#define NI_SZ  190
#define FD_SZ  320      // F*D
#define ROWS   (B_SZ * NI_SZ)   // 778240, divisible by 16

// ---------------------------------------------------------------------------
// Generic fp32 WMMA GEMM: C(M,N) = A(M,K) @ Bm(K,N), row-major, K % 4 == 0.
// One wave computes MT consecutive 16x16 M-tiles for a single N-tile,
// reusing the B fragment across the M-tiles.
// grid.x * WAVES * MT * 16 must equal M; grid.y = ceil(N/16).
// ---------------------------------------------------------------------------
#define GEMM_WAVES 4
#define GEMM_MT    2

__global__ __launch_bounds__(GEMM_WAVES * 32)
void wmma_gemm_f32(const float* __restrict__ A, const float* __restrict__ Bm,
                   float* __restrict__ C, int M, int N, int K)
{
    const int wave = threadIdx.x >> 5;
    const int lane = threadIdx.x & 31;
    const int row  = lane & 15;        // A row within tile / B-C column
    const int kh   = lane >> 4;        // K half-select (0 or 1)

    const int mtile0 = (blockIdx.x * GEMM_WAVES + wave) * GEMM_MT;
    const int ncol   = blockIdx.y * 16 + row;
    const int ncolc  = ncol < N ? ncol : (N - 1);   // clamp loads (pad tile)

    v8f acc[GEMM_MT];
#pragma unroll
    for (int m = 0; m < GEMM_MT; ++m) acc[m] = (v8f){};

    for (int k0 = 0; k0 < K; k0 += 4) {
        const int ka = k0 + 2 * kh;
        v2f bfrag;
        bfrag.x = Bm[(size_t)ka       * N + ncolc];
        bfrag.y = Bm[(size_t)(ka + 1) * N + ncolc];
#pragma unroll
        for (int m = 0; m < GEMM_MT; ++m) {
            const size_t arow = (size_t)((mtile0 + m) * 16 + row) * K + ka;
            v2f afrag = *(const v2f*)(A + arow);   // 8B-aligned: K%4==0, ka even
            acc[m] = __builtin_amdgcn_wmma_f32_16x16x4_f32(
                false, afrag, false, bfrag, (short)0, acc[m], false, false);
        }
    }

    if (ncol < N) {
#pragma unroll
        for (int m = 0; m < GEMM_MT; ++m) {
            const int rbase = (mtile0 + m) * 16 + 8 * kh;  // C: M = r + 8*(lane>>4)
#pragma unroll
            for (int r = 0; r < 8; ++r)
                C[(size_t)(rbase + r) * N + ncol] = acc[m][r];
        }
    }
}

// ---------------------------------------------------------------------------
// Gather codebook rows per (head, b, n) straight into LDS via the CDNA5
// async global->LDS path (no VGPR round trip, tracked by ASYNCcnt), then
// multiply by W_transform (32x16, staged in LDS): emb(B*NI, 16).
// 8 waves/block, one 16-row tile per wave. ROWS/16 = 48640 = 6080 blocks * 8.
// ---------------------------------------------------------------------------
__global__ __launch_bounds__(256)
void gather_transform(const float* __restrict__ codebook,
                      const int*   __restrict__ hash_idx,   // (H, B, NI)
                      const float* __restrict__ Wt,         // (32, 16)
                      float*       __restrict__ emb)        // (B*NI, 16)
{
    __shared__ __align__(16) float ldsW[32 * 16];
    __shared__ __align__(16) float ldsA[8][16 * 32];

    const int tid  = threadIdx.x;
    const int wave = tid >> 5;
    const int lane = tid & 31;
    const int r    = lane & 15;   // tile row / output column
    const int h    = lane >> 4;   // head select / K half-select

    // stage W_transform (512 floats) into LDS
    for (int t = tid; t < 32 * 16; t += 256) ldsW[t] = Wt[t];

    // each lane pulls one 64-byte codebook row straight into its LDS slot
    const long g = (long)(blockIdx.x * 8 + wave) * 16 + r;   // < ROWS exactly
    const int  b = (int)(g / NI_SZ);
    const int  n = (int)(g % NI_SZ);
    const int  cidx = hash_idx[((size_t)h * B_SZ + b) * NI_SZ + n];
    const float* src = codebook + (size_t)cidx * D_SZ;
    // LDS byte offset = low 32 bits of the generic address of the shared slot
    // (ISA 10.2); async offset applies to both sides.
    unsigned lds_off = (unsigned)(size_t)(&ldsA[wave][r * 32 + h * 16]);
    asm volatile(
        "global_load_async_to_lds_b128 %0, %1, off\n\t"
        "global_load_async_to_lds_b128 %0, %1, off offset:16\n\t"
        "global_load_async_to_lds_b128 %0, %1, off offset:32\n\t"
        "global_load_async_to_lds_b128 %0, %1, off offset:48"
        :: "v"(lds_off), "v"(src) : "memory");
    asm volatile("s_wait_asynccnt 0x0" ::: "memory");

    __syncthreads();

    v8f acc = (v8f){};
#pragma unroll
    for (int k0 = 0; k0 < 32; k0 += 4) {
        const int ka = k0 + 2 * h;
        v2f a = *(const v2f*)&ldsA[wave][r * 32 + ka];
        v2f bfrag;
        bfrag.x = ldsW[ka * 16 + r];
        bfrag.y = ldsW[(ka + 1) * 16 + r];
        acc = __builtin_amdgcn_wmma_f32_16x16x4_f32(
            false, a, false, bfrag, (short)0, acc, false, false);
    }

    const long rowbase = (long)(blockIdx.x * 8 + wave) * 16 + 8 * h;
#pragma unroll
    for (int q = 0; q < 8; ++q)
        emb[(rowbase + q) * D_SZ + r] = acc[q];
}

// ---------------------------------------------------------------------------
// out[b,i,d] = sum_{t<19} emb[b, idx[i][t], d] * w[b, idx[i][t]]
// one block per batch row; 320 threads = (i,d) pairs.
// ---------------------------------------------------------------------------
__global__ __launch_bounds__(FD_SZ)
void combine(const float* __restrict__ emb,   // (B*NI, 16)
             const float* __restrict__ wv,    // (B, NI)
             const int*   __restrict__ inter, // (F, F-1) = (20, 19)
             float*       __restrict__ out)   // (B, F, D)
{
    __shared__ float lw[192];
    __shared__ int   lidx[F_SZ * (F_SZ - 1)];   // 380

    const int b   = blockIdx.x;
    const int tid = threadIdx.x;

    if (tid < NI_SZ) lw[tid] = wv[(size_t)b * NI_SZ + tid];
    for (int t = tid; t < F_SZ * (F_SZ - 1); t += FD_SZ) lidx[t] = inter[t];
    __syncthreads();

    const int i = tid >> 4;
    const int d = tid & 15;
    const float* embB = emb + (size_t)b * NI_SZ * D_SZ;

    float acc = 0.0f;
#pragma unroll
    for (int t = 0; t < F_SZ - 1; ++t) {
        const int j = lidx[i * (F_SZ - 1) + t];
        acc += embB[j * D_SZ + d] * lw[j];
    }
    out[((size_t)b * F_SZ + i) * D_SZ + d] = acc;
}

// ---------------------------------------------------------------------------
extern "C" void kernel_launch(void* const* d_in, const int* in_sizes, int n_in,
                              void* d_out, int out_size, void* d_ws, size_t ws_size,
                              hipStream_t stream)
{
    (void)in_sizes; (void)n_in; (void)out_size; (void)ws_size;

    const float* origin   = (const float*)d_in[0];  // (B, F, D) == (B, 320)
    const float* codebook = (const float*)d_in[1];  // (NB, 16)
    const float* Wt       = (const float*)d_in[2];  // (32, 16)
    const float* w1       = (const float*)d_in[3];  // (320, 320)
    const float* w2       = (const float*)d_in[4];  // (320, 190)
    const int*   hidx     = (const int*)d_in[5];    // (2, B, 190)
    const int*   inter    = (const int*)d_in[6];    // (20, 19)
    float* out = (float*)d_out;

    float* hidden = (float*)d_ws;                       // B*320
    float* wv     = hidden + (size_t)B_SZ * FD_SZ;      // B*190
    float* emb    = wv     + (size_t)B_SZ * NI_SZ;      // B*190*16

    // hidden = z @ senet_w1   (4096 x 320 x 320)
    {
        dim3 grid(B_SZ / (16 * GEMM_MT * GEMM_WAVES), FD_SZ / 16);  // (32, 20)
        wmma_gemm_f32<<<grid, GEMM_WAVES * 32, 0, stream>>>(
            origin, w1, hidden, B_SZ, FD_SZ, FD_SZ);
    }
    // wv = hidden @ senet_w2  (4096 x 190 x 320)
    {
        dim3 grid(B_SZ / (16 * GEMM_MT * GEMM_WAVES), (NI_SZ + 15) / 16);  // (32, 12)
        wmma_gemm_f32<<<grid, GEMM_WAVES * 32, 0, stream>>>(
            hidden, w2, wv, B_SZ, NI_SZ, FD_SZ);
    }
    // emb = gather+concat @ W_transform
    gather_transform<<<ROWS / 16 / 8, 256, 0, stream>>>(codebook, hidx, Wt, emb);
    // final weighted field sum
    combine<<<B_SZ, FD_SZ, 0, stream>>>(emb, wv, inter, out);
}